// DNC_70411693851074
// MI455X (gfx1250) — compile-verified
//
#include <hip/hip_runtime.h>
#include <math.h>

#define B_     512
#define IN_    128
#define N_     256
#define W_     64
#define R_     4
#define U_     256
#define OUT_   128
#define IFACE_ 471
#define VLD_   512      // padded leading dim for the interface vector
#define CLIP_  20.0f
#define EPS_   1e-6f

typedef float v2f __attribute__((ext_vector_type(2)));
typedef float v8f __attribute__((ext_vector_type(8)));

__device__ __forceinline__ float sigm_(float x) { return 1.0f / (1.0f + expf(-x)); }
__device__ __forceinline__ float sp_(float x)   { return (x > 20.0f) ? x : log1pf(expf(x)); }

// ---------------------------------------------------------------------------
// f32 WMMA GEMM: C[M,N] = A[M,K] @ B[K,N] + bias, optional clip.
// One wave computes a 16x64 strip: 4x V_WMMA_F32_16X16X4_F32 per k-step,
// A fragment (global_load_b64) reused across all 4. No guards: M%16==0,
// K%4==0, N%64==0 (callers pad N).
// ---------------------------------------------------------------------------
__global__ void gemm_wmma_f32(const float* __restrict__ A, const float* __restrict__ Bm,
                              const float* __restrict__ bias, float* __restrict__ C,
                              int M, int N, int K, int doClip)
{
    const int lane = threadIdx.x;          // 0..31 (wave32)
    const int n0   = blockIdx.x * 64;
    const int m0   = blockIdx.y * 16;
    const int l15  = lane & 15;
    const int hi   = lane >> 4;            // 0 or 1
    const int koff = hi * 2;
    const float* arow = A + (size_t)(m0 + l15) * K;

    v8f acc0 = {}, acc1 = {}, acc2 = {}, acc3 = {};
    for (int k = 0; k < K; k += 4) {
        v2f a;
        a.x = arow[k + koff];
        a.y = arow[k + koff + 1];
        const float* b0p = Bm + (size_t)(k + koff) * N + l15;   // row kr
        const float* b1p = b0p + N;                             // row kr+1
        v2f b0, b1, b2, b3;
        b0.x = b0p[n0     ]; b0.y = b1p[n0     ];
        b1.x = b0p[n0 + 16]; b1.y = b1p[n0 + 16];
        b2.x = b0p[n0 + 32]; b2.y = b1p[n0 + 32];
        b3.x = b0p[n0 + 48]; b3.y = b1p[n0 + 48];
        acc0 = __builtin_amdgcn_wmma_f32_16x16x4_f32(false, a, false, b0, (short)0, acc0, false, false);
        acc1 = __builtin_amdgcn_wmma_f32_16x16x4_f32(false, a, false, b1, (short)0, acc1, false, false);
        acc2 = __builtin_amdgcn_wmma_f32_16x16x4_f32(false, a, false, b2, (short)0, acc2, false, false);
        acc3 = __builtin_amdgcn_wmma_f32_16x16x4_f32(false, a, false, b3, (short)0, acc3, false, false);
    }

    const int mbase = m0 + hi * 8;         // C/D layout: VGPR i -> M = i + 8*hi
#pragma unroll
    for (int t = 0; t < 4; ++t) {
        const int ncol = n0 + t * 16 + l15;
        const float bv = bias ? bias[ncol] : 0.0f;
        v8f acc = (t == 0) ? acc0 : (t == 1) ? acc1 : (t == 2) ? acc2 : acc3;
#pragma unroll
        for (int i = 0; i < 8; ++i) {
            float val = acc[i] + bv;
            if (doClip) val = fminf(fmaxf(val, -CLIP_), CLIP_);
            C[(size_t)(mbase + i) * N + ncol] = val;
        }
    }
}

// ---------------------------------------------------------------------------
// Pack x = [inputs | rv_prev.reshape(W*R) | h_prev]  -> [B, 640]
// ---------------------------------------------------------------------------
__global__ void pack_x_kernel(const float* __restrict__ inputs,
                              const float* __restrict__ rv_prev,
                              const float* __restrict__ h_prev,
                              float* __restrict__ xcat)
{
    int idx = blockIdx.x * blockDim.x + threadIdx.x;
    const int KX = IN_ + W_ * R_ + U_;     // 640
    if (idx >= B_ * KX) return;
    int b = idx / KX, j = idx % KX;
    float v;
    if (j < IN_)               v = inputs[(size_t)b * IN_ + j];
    else if (j < IN_ + W_*R_)  v = rv_prev[(size_t)b * (W_*R_) + (j - IN_)];
    else                       v = h_prev[(size_t)b * U_ + (j - IN_ - W_*R_)];
    xcat[idx] = v;
}

// Pack Wcat = [Wk ; Wr]  -> [640, 1024]
__global__ void pack_w_kernel(const float* __restrict__ Wk, const float* __restrict__ Wr,
                              float* __restrict__ Wcat)
{
    int idx = blockIdx.x * blockDim.x + threadIdx.x;
    const int KW = (IN_ + W_*R_) * 4 * U_;   // 384*1024
    const int TOT = KW + U_ * 4 * U_;        // + 256*1024
    if (idx >= TOT) return;
    Wcat[idx] = (idx < KW) ? Wk[idx] : Wr[idx - KW];
}

// Pack WiP = zero-padded Wi  [256,471] -> [256,512]
__global__ void pack_wi_kernel(const float* __restrict__ Wi, const float* __restrict__ bi,
                               float* __restrict__ WiP, float* __restrict__ biP)
{
    int idx = blockIdx.x * blockDim.x + threadIdx.x;
    if (idx < VLD_) biP[idx] = (idx < IFACE_) ? bi[idx] : 0.0f;
    if (idx >= U_ * VLD_) return;
    int row = idx / VLD_, col = idx % VLD_;
    WiP[idx] = (col < IFACE_) ? Wi[(size_t)row * IFACE_ + col] : 0.0f;
}

// ---------------------------------------------------------------------------
// LSTM gates: h = clip(sigm(zo)*tanh(sigm(zf)*c_prev + sigm(zi)*tanh(zc)))
// ---------------------------------------------------------------------------
__global__ void gates_kernel(const float* __restrict__ z, const float* __restrict__ c_prev,
                             float* __restrict__ h)
{
    int idx = blockIdx.x * blockDim.x + threadIdx.x;
    if (idx >= B_ * U_) return;
    int b = idx / U_, u = idx % U_;
    const float* zb = z + (size_t)b * 4 * U_;
    float zi = zb[u], zf = zb[U_ + u], zc = zb[2*U_ + u], zo = zb[3*U_ + u];
    float c  = sigm_(zf) * c_prev[idx] + sigm_(zi) * tanhf(zc);
    float hv = sigm_(zo) * tanhf(c);
    h[idx] = fminf(fmaxf(hv, -CLIP_), CLIP_);
}

// ---------------------------------------------------------------------------
// Usage + allocation weighting: per-batch bitonic argsort (N=256) in LDS,
// exclusive product scan, scatter back via sorted indices.
// ---------------------------------------------------------------------------
__global__ void alloc_kernel(const float* __restrict__ vif, const float* __restrict__ wr_prev,
                             const float* __restrict__ usage, const float* __restrict__ ww_prev,
                             float* __restrict__ alloc)
{
    __shared__ float key[N_];
    __shared__ int   kidx[N_];
    __shared__ float cp[N_];
    const int b = blockIdx.x, n = threadIdx.x;
    const float* vb = vif + (size_t)b * VLD_;

    float psi = 1.0f;
#pragma unroll
    for (int r = 0; r < R_; ++r) {
        float fg = sigm_(vb[453 + r]);
        psi *= (1.0f - fg * wr_prev[((size_t)b * N_ + n) * R_ + r]);
    }
    float us = usage[(size_t)b * N_ + n], wwp = ww_prev[(size_t)b * N_ + n];
    key[n]  = (us + wwp - us * wwp) * psi;
    kidx[n] = n;
    __syncthreads();

    // bitonic sort ascending (value,index)
    for (int k = 2; k <= N_; k <<= 1) {
        for (int j = k >> 1; j > 0; j >>= 1) {
            int ixj = n ^ j;
            if (ixj > n) {
                bool asc = ((n & k) == 0);
                float a = key[n], c = key[ixj];
                if ((a > c) == asc) {
                    key[n] = c; key[ixj] = a;
                    int t = kidx[n]; kidx[n] = kidx[ixj]; kidx[ixj] = t;
                }
            }
            __syncthreads();
        }
    }
    // inclusive product scan (Hillis-Steele)
    cp[n] = key[n];
    __syncthreads();
    for (int off = 1; off < N_; off <<= 1) {
        float t = (n >= off) ? cp[n - off] : 1.0f;
        __syncthreads();
        cp[n] *= t;
        __syncthreads();
    }
    float cpe = (n == 0) ? 1.0f : cp[n - 1];
    alloc[(size_t)b * N_ + kidx[n]] = (1.0f - key[n]) * cpe;
}

// ---------------------------------------------------------------------------
// Write content addressing + write weights (block softmax over N per batch).
// ---------------------------------------------------------------------------
__global__ void write_addr_kernel(const float* __restrict__ vif, const float* __restrict__ M,
                                  const float* __restrict__ alloc, float* __restrict__ ww)
{
    __shared__ float keyS[W_];
    __shared__ float red[N_];
    const int b = blockIdx.x, n = threadIdx.x;
    const float* vb = vif + (size_t)b * VLD_;
    if (n < W_) keyS[n] = vb[260 + n];
    __syncthreads();

    red[n] = (n < W_) ? keyS[n] * keyS[n] : 0.0f;
    __syncthreads();
    for (int s = 128; s > 0; s >>= 1) { if (n < s) red[n] += red[n + s]; __syncthreads(); }
    float kn = sqrtf(red[0]);
    __syncthreads();

    float wstr = 1.0f + sp_(vb[324]);
    float ag   = sigm_(vb[457]);
    float wg   = sigm_(vb[458]);

    const float* Mrow = M + ((size_t)b * N_ + n) * W_;
    float dot = 0.0f, m2 = 0.0f;
    for (int w = 0; w < W_; ++w) { float m = Mrow[w]; dot += m * keyS[w]; m2 += m * m; }
    float e = wstr * (dot / (sqrtf(m2) * kn + EPS_));

    red[n] = e; __syncthreads();
    for (int s = 128; s > 0; s >>= 1) { if (n < s) red[n] = fmaxf(red[n], red[n + s]); __syncthreads(); }
    float mx = red[0]; __syncthreads();
    float ex = expf(e - mx);
    red[n] = ex; __syncthreads();
    for (int s = 128; s > 0; s >>= 1) { if (n < s) red[n] += red[n + s]; __syncthreads(); }
    float cw = ex / red[0];

    float al = alloc[(size_t)b * N_ + n];
    ww[(size_t)b * N_ + n] = wg * (ag * al + (1.0f - ag) * cw);
}

// ---------------------------------------------------------------------------
// Fused temporal-link: fwd/bwd = L_new(@/^T@) wr_prev without materializing
// L_new (134 MB). One block per batch; thread n owns row n (fwd) + col n (bwd).
// link fits in the 192MB L2, so the second traversal is an L2 hit.
// ---------------------------------------------------------------------------
__global__ void fwdbwd_kernel(const float* __restrict__ link, const float* __restrict__ ww,
                              const float* __restrict__ prec, const float* __restrict__ wr_prev,
                              float* __restrict__ fwd, float* __restrict__ bwd)
{
    __shared__ float wwS[N_], pS[N_], wrS[N_ * R_];
    const int b = blockIdx.x, n = threadIdx.x;
    wwS[n] = ww[(size_t)b * N_ + n];
    pS[n]  = prec[(size_t)b * N_ + n];
#pragma unroll
    for (int r = 0; r < R_; ++r) wrS[n * R_ + r] = wr_prev[((size_t)b * N_ + n) * R_ + r];
    __syncthreads();

    const float* Lb = link + (size_t)b * N_ * N_;
    const float wwn = wwS[n], pn = pS[n];
    float f0 = 0, f1 = 0, f2 = 0, f3 = 0, g0 = 0, g1 = 0, g2 = 0, g3 = 0;
    for (int m = 0; m < N_; ++m) {
        if ((m & 31) == 0 && m + 32 < N_)
            __builtin_prefetch(Lb + (size_t)(m + 32) * N_ + n, 0, 1);  // global_prefetch_b8
        float lrow = Lb[(size_t)n * N_ + m];
        float lcol = Lb[(size_t)m * N_ + n];
        float coef = 1.0f - wwn - wwS[m];
        float Lr = (m == n) ? 0.0f : coef * lrow + wwn   * pS[m];
        float Lc = (m == n) ? 0.0f : coef * lcol + wwS[m] * pn;
        float w0 = wrS[m*4+0], w1 = wrS[m*4+1], w2 = wrS[m*4+2], w3 = wrS[m*4+3];
        f0 += Lr * w0; f1 += Lr * w1; f2 += Lr * w2; f3 += Lr * w3;
        g0 += Lc * w0; g1 += Lc * w1; g2 += Lc * w2; g3 += Lc * w3;
    }
    size_t o = ((size_t)b * N_ + n) * R_;
    fwd[o+0] = f0; fwd[o+1] = f1; fwd[o+2] = f2; fwd[o+3] = f3;
    bwd[o+0] = g0; bwd[o+1] = g1; bwd[o+2] = g2; bwd[o+3] = g3;
}

// ---------------------------------------------------------------------------
// Read addressing + read vectors, with M_new recomputed on the fly
// (M_new never hits HBM). Also packs cat2 = [h | read_vectors].
// ---------------------------------------------------------------------------
__global__ void read_kernel(const float* __restrict__ vif, const float* __restrict__ M,
                            const float* __restrict__ ww, const float* __restrict__ fwd,
                            const float* __restrict__ bwd, const float* __restrict__ h,
                            float* __restrict__ cat2)
{
    __shared__ float rkS[N_];             // read_keys flat: [w*4+r]
    __shared__ float krnS[R_], rsS[R_], rmS[3 * R_];
    __shared__ float erS[W_], wvS[W_];
    __shared__ float wwS[N_];
    __shared__ float red[N_];
    __shared__ float wrS[N_ * R_];
    const int b = blockIdx.x, n = threadIdx.x;
    const float* vb = vif + (size_t)b * VLD_;

    rkS[n] = vb[n];
    wwS[n] = ww[(size_t)b * N_ + n];
    if (n < W_) { erS[n] = sigm_(vb[325 + n]); wvS[n] = vb[389 + n]; }
    if (n < R_) {
        float s = 0.0f;
        for (int w = 0; w < W_; ++w) { float k = vb[w * 4 + n]; s += k * k; }
        krnS[n] = sqrtf(s);
        rsS[n]  = 1.0f + sp_(vb[256 + n]);
        float a0 = vb[459 + n], a1 = vb[463 + n], a2 = vb[467 + n];
        float mx = fmaxf(a0, fmaxf(a1, a2));
        float e0 = expf(a0 - mx), e1 = expf(a1 - mx), e2 = expf(a2 - mx);
        float s3 = e0 + e1 + e2;
        rmS[0*R_ + n] = e0 / s3; rmS[1*R_ + n] = e1 / s3; rmS[2*R_ + n] = e2 / s3;
    }
    __syncthreads();

    // per-row M_new, cosine sims against 4 keys
    const float* Mrow = M + ((size_t)b * N_ + n) * W_;
    const float wwn = wwS[n];
    float d0 = 0, d1 = 0, d2 = 0, d3 = 0, m2 = 0;
    for (int w = 0; w < W_; ++w) {
        float mn = Mrow[w] * (1.0f - wwn * erS[w]) + wwn * wvS[w];
        m2 += mn * mn;
        d0 += mn * rkS[w*4+0]; d1 += mn * rkS[w*4+1];
        d2 += mn * rkS[w*4+2]; d3 += mn * rkS[w*4+3];
    }
    float nrm = sqrtf(m2);
    float ev[R_];
    ev[0] = rsS[0] * (d0 / (nrm * krnS[0] + EPS_));
    ev[1] = rsS[1] * (d1 / (nrm * krnS[1] + EPS_));
    ev[2] = rsS[2] * (d2 / (nrm * krnS[2] + EPS_));
    ev[3] = rsS[3] * (d3 / (nrm * krnS[3] + EPS_));

    float cr[R_];
    for (int r = 0; r < R_; ++r) {
        red[n] = ev[r]; __syncthreads();
        for (int s = 128; s > 0; s >>= 1) { if (n < s) red[n] = fmaxf(red[n], red[n + s]); __syncthreads(); }
        float mx = red[0]; __syncthreads();
        float ex = expf(ev[r] - mx);
        red[n] = ex; __syncthreads();
        for (int s = 128; s > 0; s >>= 1) { if (n < s) red[n] += red[n + s]; __syncthreads(); }
        float sm = red[0]; __syncthreads();
        cr[r] = ex / sm;
    }

    size_t o = ((size_t)b * N_ + n) * R_;
#pragma unroll
    for (int r = 0; r < R_; ++r)
        wrS[n * R_ + r] = rmS[0*R_ + r] * bwd[o + r] + rmS[1*R_ + r] * cr[r] + rmS[2*R_ + r] * fwd[o + r];
    __syncthreads();

    // read_vectors[w,r] = sum_n M_new[n,w] * wr[n,r]; thread = w*4+r (256 = 64*4)
    const int w = n >> 2, r = n & 3;
    float acc = 0.0f;
    for (int m = 0; m < N_; ++m) {
        float mn = M[((size_t)b * N_ + m) * W_ + w] * (1.0f - wwS[m] * erS[w]) + wwS[m] * wvS[w];
        acc += mn * wrS[m * R_ + r];
    }
    cat2[(size_t)b * (U_ + W_*R_) + U_ + w * R_ + r] = acc;   // rv flat index w*R+r
    cat2[(size_t)b * (U_ + W_*R_) + n] = h[(size_t)b * U_ + n];
}

// ---------------------------------------------------------------------------
extern "C" void kernel_launch(void* const* d_in, const int* in_sizes, int n_in,
                              void* d_out, int out_size, void* d_ws, size_t ws_size,
                              hipStream_t stream) {
    const float* inputs  = (const float*)d_in[0];
    const float* M       = (const float*)d_in[1];
    const float* usage   = (const float*)d_in[2];
    const float* link    = (const float*)d_in[3];
    const float* prec    = (const float*)d_in[4];
    const float* ww_prev = (const float*)d_in[5];
    const float* wr_prev = (const float*)d_in[6];
    const float* h_prev  = (const float*)d_in[7];
    const float* c_prev  = (const float*)d_in[8];
    const float* rv_prev = (const float*)d_in[9];
    const float* Wk      = (const float*)d_in[10];
    const float* Wr      = (const float*)d_in[11];
    const float* b_lstm  = (const float*)d_in[12];
    const float* Wi      = (const float*)d_in[13];
    const float* bi      = (const float*)d_in[14];
    const float* Wo      = (const float*)d_in[15];
    const float* bo      = (const float*)d_in[16];
    float* out = (float*)d_out;

    const int KX = IN_ + W_*R_ + U_;   // 640
    float* ws = (float*)d_ws;
    float* Wcat  = ws;                      ws += (size_t)KX * 4 * U_;       // 640x1024
    float* xcat  = ws;                      ws += (size_t)B_ * KX;           // 512x640
    float* z     = ws;                      ws += (size_t)B_ * 4 * U_;       // 512x1024
    float* h     = ws;                      ws += (size_t)B_ * U_;           // 512x256
    float* WiP   = ws;                      ws += (size_t)U_ * VLD_;         // 256x512 padded Wi
    float* biP   = ws;                      ws += (size_t)VLD_;              // padded bi
    float* vif   = ws;                      ws += (size_t)B_ * VLD_;         // 512x512 (471 used)
    float* alloc = ws;                      ws += (size_t)B_ * N_;
    float* ww    = ws;                      ws += (size_t)B_ * N_;
    float* fwd   = ws;                      ws += (size_t)B_ * N_ * R_;
    float* bwd   = ws;                      ws += (size_t)B_ * N_ * R_;
    float* cat2  = ws;                      ws += (size_t)B_ * (U_ + W_*R_); // 512x512

    // 1) pack concat / padded operands
    {
        int tot = KX * 4 * U_;
        pack_w_kernel<<<(tot + 255) / 256, 256, 0, stream>>>(Wk, Wr, Wcat);
        tot = B_ * KX;
        pack_x_kernel<<<(tot + 255) / 256, 256, 0, stream>>>(inputs, rv_prev, h_prev, xcat);
        tot = U_ * VLD_;
        pack_wi_kernel<<<(tot + 255) / 256, 256, 0, stream>>>(Wi, bi, WiP, biP);
    }
    // 2) z = xcat @ Wcat + b_lstm   (512x640 * 640x1024), 16 x-blocks of 64 cols
    gemm_wmma_f32<<<dim3(4*U_/64, B_/16), 32, 0, stream>>>(xcat, Wcat, b_lstm, z, B_, 4*U_, KX, 0);
    // 3) LSTM gates -> h
    gates_kernel<<<(B_*U_ + 255) / 256, 256, 0, stream>>>(z, c_prev, h);
    // 4) interface v = h @ WiP + biP  (512x256 * 256x512 padded)
    gemm_wmma_f32<<<dim3(VLD_/64, B_/16), 32, 0, stream>>>(h, WiP, biP, vif, B_, VLD_, U_, 0);
    // 5) usage + allocation (sort)
    alloc_kernel<<<B_, N_, 0, stream>>>(vif, wr_prev, usage, ww_prev, alloc);
    // 6) write addressing -> ww
    write_addr_kernel<<<B_, N_, 0, stream>>>(vif, M, alloc, ww);
    // 7) fused link -> fwd/bwd
    fwdbwd_kernel<<<B_, N_, 0, stream>>>(link, ww, prec, wr_prev, fwd, bwd);
    // 8) read addressing + read vectors -> cat2 = [h | rv]
    read_kernel<<<B_, N_, 0, stream>>>(vif, M, ww, fwd, bwd, h, cat2);
    // 9) final = clip(cat2 @ Wo + bo)  (512x512 * 512x128)
    gemm_wmma_f32<<<dim3(OUT_/64, B_/16), 32, 0, stream>>>(cat2, Wo, bo, out, B_, OUT_, U_ + W_*R_, 1);
}